// DirectBEVProjector_87565793230881
// MI455X (gfx1250) — compile-verified
//
#include <hip/hip_runtime.h>
#include <stdint.h>

// ---------------------------------------------------------------------------
// BEV polar bilinear resample, MI455X (gfx1250).
// Memory-bound: ~536MB traffic -> ~23us floor @ 23.3 TB/s. Strategy:
//   K1: per-pixel transform (transcendentals exactly once per pixel)
//       -> 32B record {4 gather indices, 4 bilinear weights} in d_ws (8MB).
//   K2: per (b, row, 64-pixel tile) block covers all 256 channels; records
//       staged into LDS via the CDNA5 async-to-LDS DMA path
//       (global_load_async_to_lds_b128 + s_wait_asynccnt), then each thread
//       owns 2 adjacent pixels: 8 gathers (SGPR plane base + per-lane
//       voffsets) + one b64 store per channel step.
// ---------------------------------------------------------------------------

#define BEV_H  128
#define BEV_W  512
#define IH     128
#define IW     512
#define C_CH   256
#define B_N    4
#define JT     64      // j-pixels per block
#define TWO_PI_F 6.28318530717958647692f
#define PI_F     3.14159265358979323846f

struct __align__(16) PixRec {
    int   i0, i1, i2, i3;     // gather offsets within one [IH,IW] plane
    float w0, w1, w2, w3;     // bilinear weights (nw, ne, sw, se)
};

__device__ __forceinline__ float pmod2pi(float x) {
    float m = fmodf(x, TWO_PI_F);
    return (m < 0.0f) ? (m + TWO_PI_F) : m;   // Python-style nonneg mod
}

__global__ __launch_bounds__(256)
void bev_precompute_kernel(const float* __restrict__ rot,
                           const float* __restrict__ shift_u,
                           const float* __restrict__ shift_v,
                           const float* __restrict__ mpp,
                           PixRec* __restrict__ recs) {
    int p = blockIdx.x * 256 + threadIdx.x;           // 0 .. B*Hb*Wb-1
    if (p >= B_N * BEV_H * BEV_W) return;
    int b   = p / (BEV_H * BEV_W);
    int rem = p - b * (BEV_H * BEV_W);
    int i   = rem / BEV_W;
    int j   = rem - i * BEV_W;

    float center_v = BEV_H * 0.5f - 0.5f + shift_v[b];
    float center_u = BEV_W * 0.5f - 0.5f + shift_u[b];
    float dy = (float)i - center_v;
    float dx = (float)j - center_u;
    float radius = sqrtf(dy * dy + dx * dx);

    float theta = atan2f(dy, dx);
    theta = pmod2pi(-0.5f * PI_F + pmod2pi(theta));
    // ROT_RANGE/180*pi == 2*pi
    theta = pmod2pi(theta + rot[b] * TWO_PI_F);
    float u = theta * ((float)IW / TWO_PI_F);

    float dist = radius * mpp[0];
    float phi  = atan2f(dist, -2.0f);                 // GRD_HEIGHT = -2.0
    float v    = phi * ((float)IH / PI_F);

    float ixf = floorf(u);
    float iyf = floorf(v);
    #define CX(t) fminf(fmaxf((t), 0.0f), (float)(IW - 1))
    #define CY(t) fminf(fmaxf((t), 0.0f), (float)(IH - 1))
    float ix_ne = CX(ixf + 1.0f), iy_ne = CY(iyf);
    float ix_sw = CX(ixf),        iy_sw = CY(iyf + 1.0f);
    float ix_se = CX(ixf + 1.0f), iy_se = CY(iyf + 1.0f);
    float ix_nw = CX(ixf),        iy_nw = CY(iyf);
    #undef CX
    #undef CY

    PixRec r;
    r.w0 = (ix_se - u) * (iy_se - v);
    r.w1 = (u - ix_sw) * (iy_sw - v);
    r.w2 = (ix_ne - u) * (v - iy_ne);
    r.w3 = (u - ix_nw) * (v - iy_nw);
    r.i0 = (int)iy_nw * IW + (int)ix_nw;
    r.i1 = (int)iy_ne * IW + (int)ix_ne;
    r.i2 = (int)iy_sw * IW + (int)ix_sw;
    r.i3 = (int)iy_se * IW + (int)ix_se;
    recs[p] = r;
}

__global__ __launch_bounds__(256)
void bev_gather_kernel(const float* __restrict__ feat,
                       const PixRec* __restrict__ recs,
                       float* __restrict__ out) {
    __shared__ __align__(16) char smem[JT * sizeof(PixRec)];   // 2 KB

    const int jt  = blockIdx.x;                 // 0..7   (j tile)
    const int i   = blockIdx.y;                 // 0..127 (BEV row)
    const int b   = blockIdx.z;                 // 0..3
    const int tx  = threadIdx.x;                // 0..31  -> pixel pair
    const int ty  = threadIdx.y;                // 0..7   -> channel lane
    const int tid = ty * 32 + tx;

    // ---- Stage 64 PixRec (2KB) global -> LDS via CDNA5 async DMA path ----
    // Waves 0..3 (tid < 128): one 16B chunk per lane. The low 32 bits of a
    // flat shared pointer are the workgroup-relative LDS byte offset the
    // instruction's VDST expects.
    const unsigned long long gbase = (unsigned long long)(uintptr_t)
        (recs + ((b * BEV_H + i) * BEV_W + jt * JT));
    if (tid < 128) {
        unsigned int lds_addr =
            (unsigned int)(uintptr_t)(&smem[0] + tid * 16);
        unsigned long long gaddr = gbase + (unsigned long long)tid * 16ull;
        asm volatile("global_load_async_to_lds_b128 %0, %1, off"
                     :: "v"(lds_addr), "v"(gaddr) : "memory");
    }
    asm volatile("s_wait_asynccnt 0" ::: "memory");
    __syncthreads();

    const PixRec* rp = (const PixRec*)smem;
    const int p0 = 2 * tx, p1 = 2 * tx + 1;
    const int   a0 = rp[p0].i0, a1 = rp[p0].i1, a2 = rp[p0].i2, a3 = rp[p0].i3;
    const float x0 = rp[p0].w0, x1 = rp[p0].w1, x2 = rp[p0].w2, x3 = rp[p0].w3;
    const int   b0 = rp[p1].i0, b1 = rp[p1].i1, b2 = rp[p1].i2, b3 = rp[p1].i3;
    const float y0 = rp[p1].w0, y1 = rp[p1].w1, y2 = rp[p1].w2, y3 = rp[p1].w3;

    // plane base is wave-uniform (ty fixed within a wave) -> SGPR + voffset
    const float* plane = feat + (size_t)(b * C_CH + ty) * (IH * IW);
    float*       op    = out  + ((size_t)(b * C_CH + ty) * IH + i) * IW
                              + jt * JT + 2 * tx;
    const size_t step  = (size_t)8 * IH * IW;   // 8 channels per stride

    #pragma unroll 2
    for (int c = ty; c < C_CH; c += 8) {
        float r0 = plane[a0] * x0 + plane[a1] * x1
                 + plane[a2] * x2 + plane[a3] * x3;
        float r1 = plane[b0] * y0 + plane[b1] * y1
                 + plane[b2] * y2 + plane[b3] * y3;
        *(float2*)op = make_float2(r0, r1);     // global_store_b64, coalesced
        plane += step;
        op    += step;
    }
}

extern "C" void kernel_launch(void* const* d_in, const int* in_sizes, int n_in,
                              void* d_out, int out_size, void* d_ws, size_t ws_size,
                              hipStream_t stream) {
    const float* feat = (const float*)d_in[0];   // [4,256,128,512] f32
    const float* rot  = (const float*)d_in[1];   // [4]
    const float* su   = (const float*)d_in[2];   // [4]
    const float* sv   = (const float*)d_in[3];   // [4]
    const float* mpp  = (const float*)d_in[4];   // [1]

    PixRec* recs = (PixRec*)d_ws;                // 256K * 32B = 8MB scratch
    const int npix = B_N * BEV_H * BEV_W;        // 262144

    bev_precompute_kernel<<<(npix + 255) / 256, 256, 0, stream>>>(
        rot, su, sv, mpp, recs);

    dim3 grid(BEV_W / JT, BEV_H, B_N);           // (8, 128, 4)
    dim3 block(32, 8, 1);                        // 256 threads = 8 wave32
    bev_gather_kernel<<<grid, block, 0, stream>>>(feat, recs, (float*)d_out);
}